// BitNetEventSemanticEncoder_43576738185548
// MI455X (gfx1250) — compile-verified
//
#include <hip/hip_runtime.h>

typedef __attribute__((ext_vector_type(2))) float v2f;
typedef __attribute__((ext_vector_type(8))) float v8f;

#define D_OUT        256
#define D_IN         96
#define KSTEPS       24      // 96 / 4
#define NTILES       16      // 256 / 16
#define LDS_STRIDE   100     // 96 + 4 pad -> conflict-free ds_load_b64
#define EPS_W        1e-8f
#define EPS_LN       1e-5f

// ---------- prep: BitNet b1.58 quantization of W into d_ws ----------
// scale = mean(|W|); w_q = clip(rint(W/(scale+eps)), -1, 1) * scale
__global__ __launch_bounds__(256)
void bitnet_quant_prep(const float* __restrict__ W, float* __restrict__ wq, int n) {
    __shared__ float red[256];
    int tid = threadIdx.x;
    float s = 0.f;
    for (int i = tid; i < n; i += 256) s += fabsf(W[i]);
    red[tid] = s;
    __syncthreads();
    for (int off = 128; off > 0; off >>= 1) {
        if (tid < off) red[tid] += red[tid + off];
        __syncthreads();
    }
    float scale = red[0] / (float)n;
    float inv   = 1.0f / (scale + EPS_W);
    for (int i = tid; i < n; i += 256) {
        float q = rintf(W[i] * inv);               // round-half-to-even, like jnp.round
        q = fminf(1.0f, fmaxf(-1.0f, q));
        wq[i] = q * scale;
    }
}

// ---------- fused: gather -> WMMA f32 GEMM (96->256) -> LayerNorm ----------
__global__ __launch_bounds__(256)
void bitnet_encoder_kernel(
    const int* __restrict__ et, const int* __restrict__ fc,
    const int* __restrict__ sc, const int* __restrict__ of,
    const int* __restrict__ tt, const int* __restrict__ rc,
    const float* __restrict__ E_et, const float* __restrict__ E_fc,
    const float* __restrict__ E_sc, const float* __restrict__ E_of,
    const float* __restrict__ E_tt, const float* __restrict__ E_rc,
    const float* __restrict__ wq,  const float* __restrict__ bias,
    const float* __restrict__ gamma, const float* __restrict__ beta,
    float* __restrict__ out, int ntokens)
{
    __shared__ float sW[D_OUT * LDS_STRIDE];       // 102,400 B

    const int tid = threadIdx.x;

    // Stage quantized weights into LDS, one 96-float row per thread (16B vec).
    {
        const float4* src = (const float4*)(wq + tid * D_IN);
        #pragma unroll
        for (int k = 0; k < D_IN / 4; ++k) {
            float4 v = src[k];
            *(float4*)&sW[tid * LDS_STRIDE + 4 * k] = v;   // base tid*400B: 16B aligned
        }
    }
    __syncthreads();

    const int wave = tid >> 5;
    const int lane = tid & 31;
    const int nl   = lane & 15;    // M-row for A / N-col for B,C,D
    const int hi   = lane >> 4;    // half-wave selects K pair (A/B layout)

    const int tile    = blockIdx.x * 8 + wave;
    const int tokBase = tile * 16;
    if (tokBase >= ntokens) return;                 // wave-uniform; EXEC stays all-1s

    // ---- A fragments: 16x4 f32 per k-step. lane holds row M=nl,
    //      VGPR0/1 = K = 4j+2*hi, 4j+2*hi+1 (ISA 7.12.2 32-bit A layout).
    const int t  = tokBase + nl;
    const int i0 = et[t], i1 = fc[t], i2 = sc[t],
              i3 = of[t], i4 = tt[t], i5 = rc[t];
    const float* r0 = E_et + (size_t)i0 * 16;
    const float* r1 = E_fc + (size_t)i1 * 16;
    const float* r2 = E_sc + (size_t)i2 * 16;
    const float* r3 = E_of + (size_t)i3 * 16;
    const float* r4 = E_tt + (size_t)i4 * 16;
    const float* r5 = E_rc + (size_t)i5 * 16;
#define ROWP(c) ((c)==0?r0:(c)==1?r1:(c)==2?r2:(c)==3?r3:(c)==4?r4:r5)

    v2f a[KSTEPS];
    #pragma unroll
    for (int j = 0; j < KSTEPS; ++j) {
        const int c = (4 * j) >> 4;                // table id (k0 even: pair never straddles)
        const int e = ((4 * j) & 15) + 2 * hi;     // element within table row
        a[j] = *(const v2f*)(ROWP(c) + e);         // 8B-aligned float2 gather
    }
#undef ROWP

    // ---- Accumulators: 16 tiles of 16x16 f32; bias folded into C.
    v8f acc[NTILES];
    #pragma unroll
    for (int n = 0; n < NTILES; ++n) {
        float bv = bias[n * 16 + nl];
        acc[n] = (v8f){bv, bv, bv, bv, bv, bv, bv, bv};
    }

    // ---- GEMM: 16 n-tiles x 24 k-steps = 384 v_wmma_f32_16x16x4_f32
    const int koff = 2 * hi;
    #pragma unroll
    for (int n = 0; n < NTILES; ++n) {
        const float* wrow = &sW[(n * 16 + nl) * LDS_STRIDE + koff];
        #pragma unroll
        for (int j = 0; j < KSTEPS; ++j) {
            v2f bfrag = *(const v2f*)(wrow + 4 * j); // B: lane=col N, K pair by half-wave
            acc[n] = __builtin_amdgcn_wmma_f32_16x16x4_f32(
                false, a[j], false, bfrag, (short)0, acc[n], false, false);
        }
    }

    // ---- LayerNorm over the 256 outputs of each token.
    // C/D layout: VGPR r holds M = r + 8*hi, N = nl -> reduce over 16 lanes per half.
    float s[8], ss[8];
    #pragma unroll
    for (int r = 0; r < 8; ++r) { s[r] = 0.f; ss[r] = 0.f; }
    #pragma unroll
    for (int n = 0; n < NTILES; ++n) {
        #pragma unroll
        for (int r = 0; r < 8; ++r) {
            float v = acc[n][r];
            s[r]  += v;
            ss[r] += v * v;
        }
    }
    #pragma unroll
    for (int r = 0; r < 8; ++r) {
        #pragma unroll
        for (int m = 1; m <= 8; m <<= 1) {   // xor<=8 stays inside each 16-lane half
            s[r]  += __shfl_xor(s[r],  m, 32);
            ss[r] += __shfl_xor(ss[r], m, 32);
        }
    }
    float mu[8], rs[8];
    #pragma unroll
    for (int r = 0; r < 8; ++r) {
        float m_ = s[r] * (1.0f / 256.0f);
        float v_ = ss[r] * (1.0f / 256.0f) - m_ * m_;
        mu[r] = m_;
        rs[r] = rsqrtf(v_ + EPS_LN);
    }

    // ---- Normalize + affine + store (lanes 0..15 write 64B runs per row).
    #pragma unroll
    for (int n = 0; n < NTILES; ++n) {
        const int N = n * 16 + nl;
        const float g  = gamma[N];
        const float bt = beta[N];
        #pragma unroll
        for (int r = 0; r < 8; ++r) {
            const int tok = tokBase + r + 8 * hi;
            out[(size_t)tok * D_OUT + N] = (acc[n][r] - mu[r]) * rs[r] * g + bt;
        }
    }
}

extern "C" void kernel_launch(void* const* d_in, const int* in_sizes, int n_in,
                              void* d_out, int out_size, void* d_ws, size_t ws_size,
                              hipStream_t stream) {
    const int* et = (const int*)d_in[0];
    const int* fc = (const int*)d_in[1];
    const int* sc = (const int*)d_in[2];
    const int* of = (const int*)d_in[3];
    const int* tt = (const int*)d_in[4];
    const int* rc = (const int*)d_in[5];
    const float* E_et = (const float*)d_in[6];
    const float* E_fc = (const float*)d_in[7];
    const float* E_sc = (const float*)d_in[8];
    const float* E_of = (const float*)d_in[9];
    const float* E_tt = (const float*)d_in[10];
    const float* E_rc = (const float*)d_in[11];
    const float* W     = (const float*)d_in[12];
    const float* bias  = (const float*)d_in[13];
    const float* gamma = (const float*)d_in[14];
    const float* beta  = (const float*)d_in[15];

    float* wq = (float*)d_ws;                       // 24576 floats = 96 KB scratch
    const int nW      = in_sizes[12];               // 256*96
    const int ntokens = in_sizes[0];                // B*S = 1,048,576

    bitnet_quant_prep<<<1, 256, 0, stream>>>(W, wq, nW);

    const int tiles  = (ntokens + 15) / 16;         // 65536
    const int blocks = (tiles + 7) / 8;             // 8192 blocks x 8 waves
    bitnet_encoder_kernel<<<blocks, 256, 0, stream>>>(
        et, fc, sc, of, tt, rc,
        E_et, E_fc, E_sc, E_of, E_tt, E_rc,
        wq, bias, gamma, beta, (float*)d_out, ntokens);
}